// HTGNN_14087492731183
// MI455X (gfx1250) — compile-verified
//
#include <hip/hip_runtime.h>

#define DF 32
#define TPB 256

typedef float v2f __attribute__((ext_vector_type(2)));
typedef float v8f __attribute__((ext_vector_type(8)));

static inline int cdiv(long long a, long long b) { return (int)((a + b - 1) / b); }

__device__ __forceinline__ void atomAddF(float* p, float v) {
  unsafeAtomicAdd(p, v);  // native global_atomic_add_f32 on gfx1250
}

// Monotone float<->uint encoding so unsigned atomicMax == float max
__device__ __forceinline__ unsigned fenc(float f) {
  unsigned u = __float_as_uint(f);
  return (u & 0x80000000u) ? ~u : (u | 0x80000000u);
}
__device__ __forceinline__ float fdec(unsigned u) {
  unsigned b = (u & 0x80000000u) ? (u & 0x7FFFFFFFu) : ~u;
  return __uint_as_float(b);
}

__global__ void fill_f32(float* p, float v, int n) {
  int i = blockIdx.x * blockDim.x + threadIdx.x;
  if (i < n) p[i] = v;
}
__global__ void fill_u32(unsigned* p, unsigned v, int n) {
  int i = blockIdx.x * blockDim.x + threadIdx.x;
  if (i < n) p[i] = v;
}

// dst-degree accumulation; deg pre-filled with 1.0 (self loop)
__global__ void deg_kernel(const int* __restrict__ ei, int E, float* __restrict__ deg) {
  int e = blockIdx.x * blockDim.x + threadIdx.x;
  if (e < E) atomAddF(&deg[ei[E + e]], 1.0f);
}

__global__ void dinv_kernel(float* deg, int n) {
  int i = blockIdx.x * blockDim.x + threadIdx.x;
  if (i < n) deg[i] = rsqrtf(deg[i]);  // store deg^-1/2 in place
}

// H[n,32] = X[n,32] @ W[32,32], fp32 WMMA. One wave per 16-row tile.
// A 16x4 layout: lanes 0-15 hold K={0,1}, lanes 16-31 hold K={2,3} (2 VGPRs).
// B 4x16: VGPR j: lanes 0-15 -> K=j, lanes 16-31 -> K=2+j, N=lane%16.
// C/D 16x16: VGPR v: lanes 0-15 -> M=v, lanes 16-31 -> M=v+8, N=lane%16.
__global__ void gemm32_wmma(const float* __restrict__ X, const float* __restrict__ W,
                            float* __restrict__ H, int nrows) {
  int wave = (blockIdx.x * blockDim.x + threadIdx.x) >> 5;
  int lane = threadIdx.x & 31;
  int row0 = wave * 16;
  if (row0 >= nrows) return;           // wave-uniform: EXEC stays all-ones for WMMA
  int khalf = lane >> 4;               // which K half this lane holds
  int grow = row0 + (lane & 15);
  int ldrow = (grow < nrows) ? grow : 0;
  int nc = lane & 15;

  v8f c0 = {}; v8f c1 = {};            // output N-tiles [0,16) and [16,32)
#pragma unroll
  for (int k = 0; k < DF; k += 4) {
    v2f a;
    const float* xp = X + (size_t)ldrow * DF + k + 2 * khalf;
    a.x = xp[0]; a.y = xp[1];
    int kb = k + 2 * khalf;
    v2f b0, b1;
    b0.x = W[(kb + 0) * DF + nc];
    b0.y = W[(kb + 1) * DF + nc];
    b1.x = W[(kb + 0) * DF + 16 + nc];
    b1.y = W[(kb + 1) * DF + 16 + nc];
    c0 = __builtin_amdgcn_wmma_f32_16x16x4_f32(false, a, false, b0, (short)0, c0, false, false);
    c1 = __builtin_amdgcn_wmma_f32_16x16x4_f32(false, a, false, b1, (short)0, c1, false, false);
  }
  int mbase = row0 + 8 * khalf;
#pragma unroll
  for (int v = 0; v < 8; ++v) {
    int r = mbase + v;
    if (r < nrows) {
      H[(size_t)r * DF + nc]      = c0[v];
      H[(size_t)r * DF + 16 + nc] = c1[v];
    }
  }
}

// acc = b_gcn + b_gat + h * deg^-1  (GCN self-loop term; dinv holds deg^-1/2)
__global__ void acc_init_kernel(float* __restrict__ acc, const float* __restrict__ h,
                                const float* __restrict__ dinv,
                                const float* __restrict__ bg, const float* __restrict__ bga,
                                int total) {
  int t = blockIdx.x * blockDim.x + threadIdx.x;
  if (t >= total) return;
  int i = t >> 5, c = t & 31;
  float di = dinv[i];
  acc[t] = bg[c] + bga[c] + h[t] * di * di;
}

// GCN message scatter: 32 lanes = 32 channels of one edge
__global__ void gcn_scatter_kernel(const float* __restrict__ h, const int* __restrict__ ei,
                                   int E, const float* __restrict__ dinv,
                                   float* __restrict__ acc) {
  int t = blockIdx.x * blockDim.x + threadIdx.x;
  int e = t >> 5, c = t & 31;
  if (e >= E) return;
  int s = ei[e], d = ei[E + e];
  float coef = dinv[s] * dinv[d];
  atomAddF(&acc[(size_t)d * DF + c], h[(size_t)s * DF + c] * coef);
}

// GATv2 pass 1: e = leaky_relu(hl[src]+hr[dst], 0.2) . att ; segment max via uint atomicMax
__global__ void gat_edge_e_kernel(const float* __restrict__ hsrc, const float* __restrict__ hdst,
                                  const int* __restrict__ ei, int E,
                                  const float* __restrict__ att,
                                  float* __restrict__ ebuf, unsigned* __restrict__ menc) {
  int wid = (blockIdx.x * blockDim.x + threadIdx.x) >> 5;
  int lane = threadIdx.x & 31;
  if (wid >= E) return;
  int s = ei[wid];
  int d = ei[E + wid];
  float v = hsrc[(size_t)s * DF + lane] + hdst[(size_t)d * DF + lane];
  v = (v > 0.f) ? v : 0.2f * v;
  float p = v * att[lane];
#pragma unroll
  for (int off = 16; off > 0; off >>= 1) p += __shfl_xor(p, off);
  if (lane == 0) {
    ebuf[wid] = p;
    atomicMax(&menc[d], fenc(p));
  }
}

// decode uint-encoded max back to float in place; sentinel (no edges) -> 0 like reference
__global__ void mdec_kernel(unsigned* menc, int n) {
  int i = blockIdx.x * blockDim.x + threadIdx.x;
  if (i >= n) return;
  unsigned u = menc[i];
  float f = (u == 0u) ? 0.0f : fdec(u);
  ((float*)menc)[i] = f;
}

// GATv2 pass 2: ex = exp(e - m[dst]); sum per segment
__global__ void gat_edge_exp_kernel(float* __restrict__ ebuf, const int* __restrict__ ei, int E,
                                    const float* __restrict__ m, float* __restrict__ sbuf) {
  int e = blockIdx.x * blockDim.x + threadIdx.x;
  if (e >= E) return;
  int d = ei[E + e];
  float ex = expf(ebuf[e] - m[d]);
  ebuf[e] = ex;
  atomAddF(&sbuf[d], ex);
}

// GATv2 pass 3: acc[dst] += hl[src] * alpha
__global__ void gat_scatter_kernel(const float* __restrict__ hsrc, const int* __restrict__ ei,
                                   int E, const float* __restrict__ ebuf,
                                   const float* __restrict__ sbuf, float* __restrict__ acc) {
  int t = blockIdx.x * blockDim.x + threadIdx.x;
  int e = t >> 5, c = t & 31;
  if (e >= E) return;
  int s = ei[e], d = ei[E + e];
  float alpha = ebuf[e] / (sbuf[d] + 1e-16f);
  atomAddF(&acc[(size_t)d * DF + c], hsrc[(size_t)s * DF + c] * alpha);
}

// BatchNorm stats: per-channel sum & sumsq (block LDS reduce -> 32 atomics/block)
__global__ void bn_stats_kernel(const float* __restrict__ x, int n,
                                float* __restrict__ gsum, float* __restrict__ gsq) {
  __shared__ float ss[TPB], sq[TPB];
  int c = threadIdx.x & (DF - 1);
  int rg = threadIdx.x >> 5;  // 0..7 row groups
  float s = 0.f, q = 0.f;
  for (int r = blockIdx.x * 8 + rg; r < n; r += gridDim.x * 8) {
    float v = x[(size_t)r * DF + c];
    s += v; q += v * v;
  }
  ss[threadIdx.x] = s; sq[threadIdx.x] = q;
  __syncthreads();
  if (rg == 0) {
#pragma unroll
    for (int g = 1; g < TPB / DF; ++g) { s += ss[g * DF + c]; q += sq[g * DF + c]; }
    atomAddF(&gsum[c], s);
    atomAddF(&gsq[c], q);
  }
}

__global__ void bn_finalize_kernel(const float* gsum, const float* gsq,
                                   const float* gamma, const float* beta,
                                   float invn, float* scsh) {
  int c = threadIdx.x;
  if (c < DF) {
    float mu = gsum[c] * invn;
    float var = gsq[c] * invn - mu * mu;
    float sc = gamma[c] * rsqrtf(var + 1e-5f);
    scsh[c] = sc;
    scsh[DF + c] = beta[c] - mu * sc;
  }
}

__global__ void bn_apply_relu_kernel(const float* __restrict__ x, const float* __restrict__ scsh,
                                     float* __restrict__ y, int total) {
  int t = blockIdx.x * blockDim.x + threadIdx.x;
  if (t >= total) return;
  int c = t & 31;
  float v = x[t] * scsh[c] + scsh[DF + c];
  y[t] = (v > 0.f) ? v : 0.f;
}

// ---- host orchestration ----

static void run_side(const float* x_self, const float* x_other,
                     int Nself, int Nother, int E,
                     const int* ei_gcn, const int* ei_gat,
                     const float* Wgcn, const float* bgcn,
                     const float* Wl, const float* Wr,
                     const float* att, const float* bgat,
                     float* acc, float* tmp0, float* tmp1, float* ebuf,
                     float* deg, float* mbuf, float* sbuf, hipStream_t stream) {
  // GCN (self->self relation)
  fill_f32<<<cdiv(Nself, TPB), TPB, 0, stream>>>(deg, 1.0f, Nself);
  deg_kernel<<<cdiv(E, TPB), TPB, 0, stream>>>(ei_gcn, E, deg);
  dinv_kernel<<<cdiv(Nself, TPB), TPB, 0, stream>>>(deg, Nself);
  gemm32_wmma<<<cdiv(cdiv(Nself, 16), 8), TPB, 0, stream>>>(x_self, Wgcn, tmp0, Nself);
  acc_init_kernel<<<cdiv((long long)Nself * DF, TPB), TPB, 0, stream>>>(
      acc, tmp0, deg, bgcn, bgat, Nself * DF);
  gcn_scatter_kernel<<<cdiv((long long)E * DF, TPB), TPB, 0, stream>>>(tmp0, ei_gcn, E, deg, acc);
  // GATv2 (other->self relation)
  gemm32_wmma<<<cdiv(cdiv(Nother, 16), 8), TPB, 0, stream>>>(x_other, Wl, tmp0, Nother);
  gemm32_wmma<<<cdiv(cdiv(Nself, 16), 8), TPB, 0, stream>>>(x_self, Wr, tmp1, Nself);
  fill_u32<<<cdiv(Nself, TPB), TPB, 0, stream>>>((unsigned*)mbuf, 0u, Nself);
  fill_f32<<<cdiv(Nself, TPB), TPB, 0, stream>>>(sbuf, 0.0f, Nself);
  gat_edge_e_kernel<<<cdiv(E, 8), TPB, 0, stream>>>(tmp0, tmp1, ei_gat, E, att, ebuf,
                                                    (unsigned*)mbuf);
  mdec_kernel<<<cdiv(Nself, TPB), TPB, 0, stream>>>((unsigned*)mbuf, Nself);
  gat_edge_exp_kernel<<<cdiv(E, TPB), TPB, 0, stream>>>(ebuf, ei_gat, E, mbuf, sbuf);
  gat_scatter_kernel<<<cdiv((long long)E * DF, TPB), TPB, 0, stream>>>(tmp0, ei_gat, E, ebuf,
                                                                       sbuf, acc);
}

static void run_bn(const float* acc, const float* gamma, const float* beta,
                   float* out, int n, float* stats, hipStream_t stream) {
  fill_f32<<<1, 64, 0, stream>>>(stats, 0.0f, 2 * DF);
  int nb = cdiv(n, 8); if (nb > 1024) nb = 1024;
  bn_stats_kernel<<<nb, TPB, 0, stream>>>(acc, n, stats, stats + DF);
  bn_finalize_kernel<<<1, DF, 0, stream>>>(stats, stats + DF, gamma, beta,
                                           1.0f / (float)n, stats + 2 * DF);
  bn_apply_relu_kernel<<<cdiv((long long)n * DF, TPB), TPB, 0, stream>>>(
      acc, stats + 2 * DF, out, n * DF);
}

extern "C" void kernel_launch(void* const* d_in, const int* in_sizes, int n_in,
                              void* d_out, int out_size, void* d_ws, size_t ws_size,
                              hipStream_t stream) {
  const float* x_l       = (const float*)d_in[0];
  const float* x_h       = (const float*)d_in[1];
  const float* gcn_ll_W  = (const float*)d_in[2];
  const float* gcn_ll_b  = (const float*)d_in[3];
  const float* gcn_hh_W  = (const float*)d_in[4];
  const float* gcn_hh_b  = (const float*)d_in[5];
  const float* gat_hl_Wl = (const float*)d_in[6];
  const float* gat_hl_Wr = (const float*)d_in[7];
  const float* gat_hl_at = (const float*)d_in[8];
  const float* gat_hl_b  = (const float*)d_in[9];
  const float* gat_lh_Wl = (const float*)d_in[10];
  const float* gat_lh_Wr = (const float*)d_in[11];
  const float* gat_lh_at = (const float*)d_in[12];
  const float* gat_lh_b  = (const float*)d_in[13];
  const float* bn_l_g    = (const float*)d_in[14];
  const float* bn_l_b    = (const float*)d_in[15];
  const float* bn_h_g    = (const float*)d_in[16];
  const float* bn_h_b    = (const float*)d_in[17];
  const int*   ei_ll     = (const int*)d_in[18];
  const int*   ei_hh     = (const int*)d_in[19];
  const int*   ei_hl     = (const int*)d_in[20];
  const int*   ei_lh     = (const int*)d_in[21];

  int Nl = in_sizes[0] / DF;
  int Nh = in_sizes[1] / DF;
  int E  = in_sizes[18] / 2;
  int NM = (Nl > Nh) ? Nl : Nh;
  size_t NB = (size_t)NM * DF;

  float* ws    = (float*)d_ws;
  float* tmp0  = ws;            // projected features (GCN h / GAT hl_src)
  float* tmp1  = tmp0 + NB;     // GAT hr_dst
  float* accl  = tmp1 + NB;     // low-node accumulator
  float* acch  = accl + NB;     // high-node accumulator
  float* xlb   = acch + NB;     // intermediate layer output (low)
  float* xhb   = xlb + NB;      // intermediate layer output (high)
  float* ebuf  = xhb + NB;      // per-edge logits / exp values
  float* deg   = ebuf + E;      // degree -> deg^-1/2
  float* mbuf  = deg + NM;      // segment max (uint-encoded then float)
  float* sbuf  = mbuf + NM;     // segment softmax denominator
  float* stats = sbuf + NM;     // 32 sum + 32 sumsq + 64 scale/shift

  const float* curl = x_l;
  const float* curh = x_h;
  float* out_l = (float*)d_out;
  float* out_h = out_l + (size_t)Nl * DF;

  for (int layer = 0; layer < 2; ++layer) {
    size_t wo = (size_t)layer * DF * DF;
    size_t vo = (size_t)layer * DF;
    float* nl = (layer == 1) ? out_l : xlb;
    float* nh = (layer == 1) ? out_h : xhb;
    // low-frequency target: GCN(ll) + GATv2(hl)
    run_side(curl, curh, Nl, Nh, E, ei_ll, ei_hl,
             gcn_ll_W + wo, gcn_ll_b + vo,
             gat_hl_Wl + wo, gat_hl_Wr + wo, gat_hl_at + vo, gat_hl_b + vo,
             accl, tmp0, tmp1, ebuf, deg, mbuf, sbuf, stream);
    // high-frequency target: GCN(hh) + GATv2(lh)
    run_side(curh, curl, Nh, Nl, E, ei_hh, ei_lh,
             gcn_hh_W + wo, gcn_hh_b + vo,
             gat_lh_Wl + wo, gat_lh_Wr + wo, gat_lh_at + vo, gat_lh_b + vo,
             acch, tmp0, tmp1, ebuf, deg, mbuf, sbuf, stream);
    run_bn(accl, bn_l_g + vo, bn_l_b + vo, nl, Nl, stats, stream);
    run_bn(acch, bn_h_g + vo, bn_h_b + vo, nh, Nh, stats, stream);
    curl = nl; curh = nh;
  }
}